// max_deco_50723563765836
// MI455X (gfx1250) — compile-verified
//
#include <hip/hip_runtime.h>
#include <math.h>

// Reference: z (100000,256) f32; edge_index (2,m) int; out (m,) f32
//   p[m,k] = sum_d z[src[m]][k*32+d] * z[trg[m]][k*32+d],  k<8, d<32
//   out[m] = sigmoid(max_k p[m,k])
//
// Strategy: 16 edges per wave32. For each capsule k, chain 8x
// V_WMMA_F32_16X16X4_F32 with B = all-ones so the matrix pipe performs the
// 32-element dot-product reduction (row-sums of the elementwise-product
// matrix). A-matrix f32 16x4 layout (ISA 7.12.2): lane L holds row M=L&15;
// lanes 0-15 supply K=0,1 (VGPR0,VGPR1), lanes 16-31 supply K=2,3.
// D layout: VGPR r holds edge (base+r) for lanes 0-15 and (base+8+r) for
// lanes 16-31, replicated across columns -> max/sigmoid are lane-local,
// lanes 0 and 16 store 8 contiguous f32 each (two b128 stores).

typedef __attribute__((ext_vector_type(2))) float v2f;
typedef __attribute__((ext_vector_type(8))) float v8f;

#define CAPS 8
#define CAPD 32
#define ROWF 256   // floats per z row

__global__ __launch_bounds__(256) void edge_caps_wmma_kernel(
    const float* __restrict__ z,
    const int* __restrict__ ei,   // [2, mTotal] flattened
    float* __restrict__ out,
    int mTotal, int nWaves)
{
    const int lane  = threadIdx.x & 31;
    const int wave  = blockIdx.x * (blockDim.x >> 5) + (threadIdx.x >> 5);
    if (wave >= nWaves) return;          // wave-uniform: EXEC stays all-1s for WMMA

    const int base = wave * 16;
    const int m    = lane & 15;          // edge-in-tile == A-matrix row M
    const int h    = lane >> 4;          // half: supplies K=2h,2h+1 of each chunk

    const int rs = ei[base + m];
    const int rt = ei[mTotal + base + m];
    const float* __restrict__ ps = z + (size_t)rs * ROWF + h * 2;
    const float* __restrict__ pt = z + (size_t)rt * ROWF + h * 2;

    v2f b; b.x = 1.0f; b.y = 1.0f;       // ones B(4x16): layout-independent
    v8f vmax;

    #pragma unroll
    for (int k = 0; k < CAPS; ++k) {
        v8f acc = {};
        #pragma unroll
        for (int c = 0; c < CAPD / 4; ++c) {
            const int off = k * CAPD + c * 4;      // +2h folded into ps/pt
            v2f s = *(const v2f*)(ps + off);
            v2f t = *(const v2f*)(pt + off);
            v2f a; a.x = s.x * t.x; a.y = s.y * t.y;
            // D = A(16x4) * ones(4x16) + C  -> accumulate row sums over d
            acc = __builtin_amdgcn_wmma_f32_16x16x4_f32(
                false, a, false, b, (short)0, acc, false, false);
        }
        if (k == 0) {
            vmax = acc;
        } else {
            #pragma unroll
            for (int r = 0; r < 8; ++r) vmax[r] = fmaxf(vmax[r], acc[r]);
        }
    }

    #pragma unroll
    for (int r = 0; r < 8; ++r) vmax[r] = 1.0f / (1.0f + __expf(-vmax[r]));

    if (m == 0) {                        // lanes 0 and 16: 8 edges each
        float* o = out + base + h * 8;   // 32B-aligned (base mult of 16)
        *(float4*)(o)     = make_float4(vmax[0], vmax[1], vmax[2], vmax[3]);
        *(float4*)(o + 4) = make_float4(vmax[4], vmax[5], vmax[6], vmax[7]);
    }
}

// Scalar tail for mTotal % 16 != 0 (not launched for the 500000-edge case).
__global__ void edge_caps_tail_kernel(
    const float* __restrict__ z, const int* __restrict__ ei,
    float* __restrict__ out, int mStart, int mTotal)
{
    int e = mStart + blockIdx.x * blockDim.x + threadIdx.x;
    if (e >= mTotal) return;
    const float* ps = z + (size_t)ei[e] * ROWF;
    const float* pt = z + (size_t)ei[mTotal + e] * ROWF;
    float best = -INFINITY;
    for (int k = 0; k < CAPS; ++k) {
        float s = 0.0f;
        for (int d = 0; d < CAPD; ++d)
            s += ps[k * CAPD + d] * pt[k * CAPD + d];
        best = fmaxf(best, s);
    }
    out[e] = 1.0f / (1.0f + __expf(-best));
}

extern "C" void kernel_launch(void* const* d_in, const int* in_sizes, int n_in,
                              void* d_out, int out_size, void* d_ws, size_t ws_size,
                              hipStream_t stream) {
    (void)n_in; (void)d_ws; (void)ws_size;
    const float* z  = (const float*)d_in[0];
    const int*   ei = (const int*)d_in[1];
    float*       out = (float*)d_out;

    const int mTotal = in_sizes[1] / 2;          // 500000
    const int nWaves = mTotal / 16;              // 16 edges per wave
    const int wavesPerBlock = 8;                 // 256 threads
    if (nWaves > 0) {
        int blocks = (nWaves + wavesPerBlock - 1) / wavesPerBlock;
        edge_caps_wmma_kernel<<<blocks, wavesPerBlock * 32, 0, stream>>>(
            z, ei, out, mTotal, nWaves);
    }
    const int tailStart = nWaves * 16;
    const int tail = mTotal - tailStart;
    if (tail > 0) {
        int blocks = (tail + 255) / 256;
        edge_caps_tail_kernel<<<blocks, 256, 0, stream>>>(z, ei, out, tailStart, mTotal);
    }
    (void)out_size;
}